// GAT_69337952026912
// MI455X (gfx1250) — compile-verified
//
#include <hip/hip_runtime.h>
#include <cmath>

typedef float v2f __attribute__((ext_vector_type(2)));
typedef float v8f __attribute__((ext_vector_type(8)));

#define GAT_HEADS 4
#define NEG_SLOPE 0.2f

// order-preserving float <-> uint map for atomicMax-based segment max
__device__ __forceinline__ unsigned encf(float x) {
  unsigned u = __float_as_uint(x);
  return (u & 0x80000000u) ? ~u : (u | 0x80000000u);
}
__device__ __forceinline__ float decf(unsigned u) {
  u = (u & 0x80000000u) ? (u & 0x7FFFFFFFu) : ~u;
  return __uint_as_float(u);
}

// ---------------- edge_attr column means (for self-loop fill) ----------------
__global__ void colmean_kernel(const float* __restrict__ ea, int E, int ED,
                               float* __restrict__ out) {
  __shared__ float red[256];
  int d = blockIdx.x;
  float s = 0.f;
  for (int e = threadIdx.x; e < E; e += blockDim.x) s += ea[(size_t)e * ED + d];
  red[threadIdx.x] = s;
  __syncthreads();
  for (int off = 128; off > 0; off >>= 1) {
    if ((int)threadIdx.x < off) red[threadIdx.x] += red[threadIdx.x + off];
    __syncthreads();
  }
  if (threadIdx.x == 0) out[d] = red[0] / (float)E;
}

// ---------------- fp32 WMMA GEMM: C[M,Nc] = act(A[M,K] @ B[K,Nc] + bias) ------
// One wave per block; each wave computes a 16x64 tile via 4 accumulators.
// Optional split-K A operand: for k >= splitK, A element comes from
// A2[gidx[m]*lda2 + (k - splitK)]  (used to fuse concat([h, up[batch]])).
__global__ void gemm_wmma_kernel(const float* __restrict__ A, int lda,
                                 const float* __restrict__ A2, int lda2, int splitK,
                                 const int* __restrict__ gidx,
                                 const float* __restrict__ B, int ldb,
                                 const float* __restrict__ bias,
                                 float* __restrict__ C, int ldc,
                                 int K, int act) {
  const int lane = threadIdx.x;                 // 0..31
  const int m    = blockIdx.x * 16 + (lane & 15);
  const int klo  = (lane >> 4) << 1;            // 0 or 2
  const int n0   = blockIdx.y * 64 + (lane & 15);
  const size_t arow = (size_t)m * lda;
  size_t arow2 = 0;
  if (splitK < K) arow2 = (size_t)gidx[m] * lda2;

  v8f acc[4];
#pragma unroll
  for (int t = 0; t < 4; ++t) acc[t] = {};

  for (int k = 0; k < K; k += 4) {
    const int k0 = k + klo, k1 = k0 + 1;
    float a0 = (k0 < splitK) ? A[arow + k0] : A2[arow2 + (k0 - splitK)];
    float a1 = (k1 < splitK) ? A[arow + k1] : A2[arow2 + (k1 - splitK)];
    v2f av = {a0, a1};
    const float* Brow0 = B + (size_t)k0 * ldb + n0;
    const float* Brow1 = B + (size_t)k1 * ldb + n0;
#pragma unroll
    for (int t = 0; t < 4; ++t) {
      v2f bv = {Brow0[t * 16], Brow1[t * 16]};
      acc[t] = __builtin_amdgcn_wmma_f32_16x16x4_f32(false, av, false, bv,
                                                     (short)0, acc[t], false, false);
    }
  }

  const int rbase = blockIdx.x * 16 + ((lane >> 4) << 3);
  const int cb    = blockIdx.y * 64 + (lane & 15);
#pragma unroll
  for (int t = 0; t < 4; ++t) {
#pragma unroll
    for (int j = 0; j < 8; ++j) {
      int r = rbase + j;
      int c = cb + t * 16;
      float v = acc[t][j];
      if (bias) v += bias[c];
      if (act == 1) v = fmaxf(v, 0.f);
      C[(size_t)r * ldc + c] = v;
    }
  }
}

// ---------------- per-node attention logits from hp ---------------------------
__global__ void al_nodes_kernel(const float* __restrict__ hp,
                                const float* __restrict__ a_src,
                                const float* __restrict__ a_dst,
                                int C, int HC, int N,
                                float* __restrict__ alsrc, float* __restrict__ aldst) {
  int idx = blockIdx.x * blockDim.x + threadIdx.x;
  if (idx >= N * GAT_HEADS) return;
  int n = idx >> 2, hh = idx & 3;
  const float* row = hp + (size_t)n * HC + hh * C;
  const float* as = a_src + hh * C;
  const float* ad = a_dst + hh * C;
  float ss = 0.f, sd = 0.f;
  for (int c = 0; c < C; ++c) {
    float v = row[c];
    ss += v * as[c];
    sd += v * ad[c];
  }
  alsrc[idx] = ss;
  aldst[idx] = sd;
}

// ve[d][h] = sum_c We[d][h*C+c] * a_edge[h][c]   (16x4 reduced edge projection)
__global__ void ve_kernel(const float* __restrict__ We, const float* __restrict__ ae,
                          int C, float* __restrict__ ve) {
  int tid = threadIdx.x;
  if (tid >= 16 * GAT_HEADS) return;
  int d = tid >> 2, hh = tid & 3;
  int HC = GAT_HEADS * C;
  const float* w = We + (size_t)d * HC + hh * C;
  const float* a = ae + hh * C;
  float s = 0.f;
  for (int c = 0; c < C; ++c) s += w[c] * a[c];
  ve[d * GAT_HEADS + hh] = s;
}

// self-loop edge logit (same for all loops): alel[h] = ea_mean . ve[:,h]
__global__ void alel_kernel(const float* __restrict__ eamean,
                            const float* __restrict__ ve, float* __restrict__ alel) {
  int hh = threadIdx.x;
  if (hh >= GAT_HEADS) return;
  float s = 0.f;
  for (int d = 0; d < 16; ++d) s += eamean[d] * ve[d * GAT_HEADS + hh];
  alel[hh] = s;
}

// al_edge[e][h] = edge_attr[e] . ve[:,h]
__global__ void al_edge_kernel(const float* __restrict__ ea,
                               const float* __restrict__ ve, int E,
                               float* __restrict__ ale) {
  int e = blockIdx.x * blockDim.x + threadIdx.x;
  if (e >= E) return;
  const float* row = ea + (size_t)e * 16;
  float s0 = 0, s1 = 0, s2 = 0, s3 = 0;
#pragma unroll
  for (int d = 0; d < 16; ++d) {
    float v = row[d];
    const float* vv = ve + d * 4;
    s0 += v * vv[0]; s1 += v * vv[1]; s2 += v * vv[2]; s3 += v * vv[3];
  }
  float* o = ale + (size_t)e * 4;
  o[0] = s0; o[1] = s1; o[2] = s2; o[3] = s3;
}

// leaky-relu logit + segment max (encoded u32 atomicMax)
__global__ void alpha_max_kernel(const int* __restrict__ src, const int* __restrict__ dst,
                                 const float* __restrict__ alsrc, const float* __restrict__ aldst,
                                 const float* __restrict__ ale, const float* __restrict__ alel,
                                 int E, int Etot,
                                 float* __restrict__ alpha, unsigned* __restrict__ mEnc) {
  int idx = blockIdx.x * blockDim.x + threadIdx.x;
  if (idx >= Etot * GAT_HEADS) return;
  int e = idx >> 2, hh = idx & 3;
  int s, d;
  float ae;
  if (e < E) { s = src[e]; d = dst[e]; ae = ale[(size_t)e * 4 + hh]; }
  else       { s = d = e - E;          ae = alel[hh]; }
  float a = alsrc[s * 4 + hh] + aldst[d * 4 + hh] + ae;
  a = (a > 0.f) ? a : NEG_SLOPE * a;
  alpha[(size_t)e * 4 + hh] = a;
  atomicMax(&mEnc[d * 4 + hh], encf(a));
}

// p = exp(alpha - max); segment sum
__global__ void exp_sum_kernel(const int* __restrict__ dst, int E, int Etot,
                               float* __restrict__ alpha, const unsigned* __restrict__ mEnc,
                               float* __restrict__ ssum) {
  int idx = blockIdx.x * blockDim.x + threadIdx.x;
  if (idx >= Etot * GAT_HEADS) return;
  int e = idx >> 2, hh = idx & 3;
  int d = (e < E) ? dst[e] : e - E;
  float mm = decf(mEnc[d * 4 + hh]);
  float p = expf(alpha[(size_t)e * 4 + hh] - mm);
  alpha[(size_t)e * 4 + hh] = p;
  atomicAdd(&ssum[d * 4 + hh], p);
}

// wave-per-edge weighted scatter: acc[dst] += (p/s) * hp[src]
__global__ void aggregate_kernel(const int* __restrict__ src, const int* __restrict__ dst,
                                 const float* __restrict__ p, const float* __restrict__ ssum,
                                 const float* __restrict__ hp, float* __restrict__ acc,
                                 int E, int Etot, int HC, int C) {
  int gtid = blockIdx.x * blockDim.x + threadIdx.x;
  int e = gtid >> 5;
  int lane = gtid & 31;
  if (e >= Etot) return;
  int s, d;
  if (e < E) { s = src[e]; d = dst[e]; } else { s = d = e - E; }
  const float* hrow = hp + (size_t)s * HC;
  float* arow = acc + (size_t)d * HC;
  const float* pe = p + (size_t)e * 4;
  const float* sd = ssum + (size_t)d * 4;
  for (int j = 0; j * 32 < HC; ++j) {
    int ch = j * 32 + lane;
    int hh = ch / C;
    float coef = pe[hh] / sd[hh];
    atomicAdd(&arow[ch], coef * hrow[ch]);
  }
}

// concat layers: h = elu(acc + b) (in place over h buffer)
__global__ void finalize_concat_kernel(const float* __restrict__ acc,
                                       const float* __restrict__ b,
                                       float* __restrict__ h, int N) {
  int idx = blockIdx.x * blockDim.x + threadIdx.x;
  if (idx >= N * 128) return;
  int ch = idx & 127;
  float v = acc[idx] + b[ch];
  h[idx] = (v > 0.f) ? v : expm1f(v);
}

// last layer (concat=False): h = elu(mean_heads(acc) + b)
__global__ void finalize_mean_kernel(const float* __restrict__ acc,
                                     const float* __restrict__ b,
                                     float* __restrict__ hout, int N) {
  int idx = blockIdx.x * blockDim.x + threadIdx.x;
  if (idx >= N * 128) return;
  int n = idx >> 7, c = idx & 127;
  const float* a = acc + (size_t)n * 512;
  float v = 0.25f * (a[c] + a[128 + c] + a[256 + c] + a[384 + c]) + b[c];
  hout[idx] = (v > 0.f) ? v : expm1f(v);
}

// final 64 -> 1 projection
__global__ void sp3_kernel(const float* __restrict__ z, const float* __restrict__ W,
                           const float* __restrict__ b, float* __restrict__ out, int N) {
  int n = blockIdx.x * blockDim.x + threadIdx.x;
  if (n >= N) return;
  const float* r = z + (size_t)n * 64;
  float s = 0.f;
#pragma unroll
  for (int k = 0; k < 64; ++k) s += r[k] * W[k];
  out[n] = s + b[0];
}

extern "C" void kernel_launch(void* const* d_in, const int* in_sizes, int n_in,
                              void* d_out, int out_size, void* d_ws, size_t ws_size,
                              hipStream_t stream) {
  (void)n_in; (void)out_size; (void)ws_size;
  const float* x       = (const float*)d_in[0];
  const int*   eidx    = (const int*)d_in[1];
  const float* eattr   = (const float*)d_in[2];
  const float* u       = (const float*)d_in[3];
  const int*   batch   = (const int*)d_in[4];
  const float* node_W  = (const float*)d_in[5];
  const float* node_b  = (const float*)d_in[6];
  const float* gW      = (const float*)d_in[7];
  const float* gb      = (const float*)d_in[8];
  const float* ga_src  = (const float*)d_in[9];
  const float* ga_dst  = (const float*)d_in[10];
  const float* ga_edge = (const float*)d_in[11];
  const float* gWe     = (const float*)d_in[12];
  const float* lW      = (const float*)d_in[13];
  const float* lb      = (const float*)d_in[14];
  const float* la_src  = (const float*)d_in[15];
  const float* la_dst  = (const float*)d_in[16];
  const float* la_edge = (const float*)d_in[17];
  const float* lWe     = (const float*)d_in[18];
  const float* glob_W1 = (const float*)d_in[19];
  const float* glob_b1 = (const float*)d_in[20];
  const float* glob_W2 = (const float*)d_in[21];
  const float* glob_b2 = (const float*)d_in[22];
  const float* sp_W1   = (const float*)d_in[23];
  const float* sp_b1   = (const float*)d_in[24];
  const float* sp_W2   = (const float*)d_in[25];
  const float* sp_b2   = (const float*)d_in[26];
  const float* sp_W3   = (const float*)d_in[27];
  const float* sp_b3   = (const float*)d_in[28];

  const int N = in_sizes[0] / 64;        // 50000 (multiple of 16)
  const int E = in_sizes[1] / 2;         // 800000
  const int Etot = E + N;                // with self-loops
  const int* src = eidx;
  const int* dst = eidx + E;

  // ---- workspace layout (floats) ----
  float* ws = (float*)d_ws;
  size_t o = 0;
  float* h      = ws + o; o += (size_t)N * 128;   // node features (ping, reused in place)
  float* hp     = ws + o; o += (size_t)N * 512;   // transformed features (max layer)
  float* acc    = ws + o; o += (size_t)N * 512;   // aggregation accumulator
  float* alsrc  = ws + o; o += (size_t)N * 4;
  float* aldst  = ws + o; o += (size_t)N * 4;
  unsigned* mEnc = (unsigned*)(ws + o); o += (size_t)N * 4;
  float* ssum   = ws + o; o += (size_t)N * 4;
  float* ale    = ws + o; o += (size_t)E * 4;
  float* alpha  = ws + o; o += (size_t)Etot * 4;
  float* eamean = ws + o; o += 16;
  float* ve     = ws + o; o += 64;
  float* alel   = ws + o; o += 4;
  float* utmp   = ws + o; o += 64 * 128;
  float* z1     = ws + o; o += (size_t)N * 64;
  float* z2     = ws + o; o += (size_t)N * 64;

  // ---- output layout: shifts(N) | h(N*128) | edge_attr(E*16) | up(64*128) ----
  float* out_shift = (float*)d_out;
  float* out_h  = out_shift + N;
  float* out_ea = out_h + (size_t)N * 128;
  float* out_up = out_ea + (size_t)E * 16;

  const int NOSPLIT = 1 << 30;
  dim3 b32(32);
  const int eT = (Etot * GAT_HEADS + 255) / 256;
  const int eB = (E + 255) / 256;
  const int wavesB = (Etot + 7) / 8;

  // edge_attr column means (self-loop fill value)
  colmean_kernel<<<16, 256, 0, stream>>>(eattr, E, 16, eamean);

  // node embedding: h = x @ node_W + node_b
  gemm_wmma_kernel<<<dim3(N / 16, 2), b32, 0, stream>>>(
      x, 64, x, 64, NOSPLIT, nullptr, node_W, 128, node_b, h, 128, 64, 0);

  // global MLP: up = relu(u@W1+b1)@W2+b2  (written straight into d_out tail)
  gemm_wmma_kernel<<<dim3(4, 2), b32, 0, stream>>>(
      u, 32, u, 32, NOSPLIT, nullptr, glob_W1, 128, glob_b1, utmp, 128, 32, 1);
  gemm_wmma_kernel<<<dim3(4, 2), b32, 0, stream>>>(
      utmp, 128, utmp, 128, NOSPLIT, nullptr, glob_W2, 128, glob_b2, out_up, 128, 128, 0);

  // ---- 3 concat GAT layers ----
  for (int i = 0; i < 3; ++i) {
    const float* Wi  = gW + (size_t)i * 128 * 128;
    const float* bi  = gb + (size_t)i * 128;
    const float* asr = ga_src + (size_t)i * GAT_HEADS * 32;
    const float* ads = ga_dst + (size_t)i * GAT_HEADS * 32;
    const float* aed = ga_edge + (size_t)i * GAT_HEADS * 32;
    const float* Wei = gWe + (size_t)i * 16 * 128;

    gemm_wmma_kernel<<<dim3(N / 16, 2), b32, 0, stream>>>(
        h, 128, h, 128, NOSPLIT, nullptr, Wi, 128, nullptr, hp, 128, 128, 0);
    al_nodes_kernel<<<(N * GAT_HEADS + 255) / 256, 256, 0, stream>>>(
        hp, asr, ads, 32, 128, N, alsrc, aldst);
    ve_kernel<<<1, 64, 0, stream>>>(Wei, aed, 32, ve);
    alel_kernel<<<1, 4, 0, stream>>>(eamean, ve, alel);
    al_edge_kernel<<<eB, 256, 0, stream>>>(eattr, ve, E, ale);

    hipMemsetAsync(mEnc, 0, (size_t)N * 4 * sizeof(unsigned), stream);
    hipMemsetAsync(ssum, 0, (size_t)N * 4 * sizeof(float), stream);
    hipMemsetAsync(acc, 0, (size_t)N * 128 * sizeof(float), stream);

    alpha_max_kernel<<<eT, 256, 0, stream>>>(src, dst, alsrc, aldst, ale, alel,
                                             E, Etot, alpha, mEnc);
    exp_sum_kernel<<<eT, 256, 0, stream>>>(dst, E, Etot, alpha, mEnc, ssum);
    aggregate_kernel<<<wavesB, 256, 0, stream>>>(src, dst, alpha, ssum, hp, acc,
                                                 E, Etot, 128, 32);
    finalize_concat_kernel<<<(N * 128 + 255) / 256, 256, 0, stream>>>(acc, bi, h, N);
  }

  // ---- last GAT layer (concat=False, out per head = 128) ----
  gemm_wmma_kernel<<<dim3(N / 16, 8), b32, 0, stream>>>(
      h, 128, h, 128, NOSPLIT, nullptr, lW, 512, nullptr, hp, 512, 128, 0);
  al_nodes_kernel<<<(N * GAT_HEADS + 255) / 256, 256, 0, stream>>>(
      hp, la_src, la_dst, 128, 512, N, alsrc, aldst);
  ve_kernel<<<1, 64, 0, stream>>>(lWe, la_edge, 128, ve);
  alel_kernel<<<1, 4, 0, stream>>>(eamean, ve, alel);
  al_edge_kernel<<<eB, 256, 0, stream>>>(eattr, ve, E, ale);

  hipMemsetAsync(mEnc, 0, (size_t)N * 4 * sizeof(unsigned), stream);
  hipMemsetAsync(ssum, 0, (size_t)N * 4 * sizeof(float), stream);
  hipMemsetAsync(acc, 0, (size_t)N * 512 * sizeof(float), stream);

  alpha_max_kernel<<<eT, 256, 0, stream>>>(src, dst, alsrc, aldst, ale, alel,
                                           E, Etot, alpha, mEnc);
  exp_sum_kernel<<<eT, 256, 0, stream>>>(dst, E, Etot, alpha, mEnc, ssum);
  aggregate_kernel<<<wavesB, 256, 0, stream>>>(src, dst, alpha, ssum, hp, acc,
                                               E, Etot, 512, 128);
  finalize_mean_kernel<<<(N * 128 + 255) / 256, 256, 0, stream>>>(acc, lb, out_h, N);

  // passthrough output: edge_attr
  hipMemcpyAsync(out_ea, eattr, (size_t)E * 16 * sizeof(float),
                 hipMemcpyDeviceToDevice, stream);

  // ---- shift predictor MLP; concat([h, up[batch]]) fused via split-K A fetch ----
  gemm_wmma_kernel<<<dim3(N / 16, 1), b32, 0, stream>>>(
      out_h, 128, out_up, 128, /*splitK=*/128, batch, sp_W1, 64, sp_b1, z1, 64, 256, 1);
  gemm_wmma_kernel<<<dim3(N / 16, 1), b32, 0, stream>>>(
      z1, 64, z1, 64, NOSPLIT, nullptr, sp_W2, 64, sp_b2, z2, 64, 64, 1);
  sp3_kernel<<<(N + 255) / 256, 256, 0, stream>>>(z2, sp_W3, sp_b3, out_shift, N);
}